// SymmetricHermiteBlock_73847667687667
// MI455X (gfx1250) — compile-verified
//
#include <hip/hip_runtime.h>
#include <hip/hip_bf16.h>
#include <math.h>

typedef float v2f __attribute__((ext_vector_type(2)));
typedef float v8f __attribute__((ext_vector_type(8)));

#define BATCH 4096
#define DDIM  128
#define HDIM  256
#define HT    768      // 3 maps * 256
#define AROWS 512      // rows from A_w (2*256); rows >= AROWS come from B_w (sign -1)

// Stacked weight row j: rows [0,512) from A_w, rows [512,768) from B_w.
static __device__ __forceinline__ const float* wall_row(const float* __restrict__ A,
                                                        const float* __restrict__ B, int j) {
    return (j < AROWS) ? (A + (size_t)j * DDIM) : (B + (size_t)(j - AROWS) * DDIM);
}

// D = A(16x4, f32) * B(4x16, f32) + C  — exact fp32 on the matrix pipe.
static __device__ __forceinline__ v8f wmma_f32(v2f a, v2f b, v8f c) {
    return __builtin_amdgcn_wmma_f32_16x16x4_f32(false, a, false, b, (short)0, c, false, false);
}

// ---------------------------------------------------------------------------
// M[i,j] = sign_i * sign_j * (W_i . W_j)^2   (768x768, symmetric,
// batch-independent).  One wave per 16x16 tile; K = 128 in steps of 4.
// ---------------------------------------------------------------------------
__global__ void k_ms(const float* __restrict__ A, const float* __restrict__ B,
                     float* __restrict__ Ms) {
    const int lane = threadIdx.x & 31;
    const int l15  = lane & 15;
    const int hi   = lane >> 4;
    const int i0   = blockIdx.x * 16;
    const int j0   = blockIdx.y * 16;
    const float* ra = wall_row(A, B, i0 + l15);  // A-matrix row m = l15
    const float* rb = wall_row(A, B, j0 + l15);  // B-matrix col n = l15 (B = W^T)
    v8f acc = {};
#pragma unroll 8
    for (int k0 = 0; k0 < DDIM; k0 += 4) {
        v2f a = *reinterpret_cast<const v2f*>(ra + k0 + 2 * hi);
        v2f b = *reinterpret_cast<const v2f*>(rb + k0 + 2 * hi);
        acc = wmma_f32(a, b, acc);
    }
    const float ss = ((i0 < AROWS) ? 1.f : -1.f) * ((j0 < AROWS) ? 1.f : -1.f);
#pragma unroll
    for (int g = 0; g < 8; ++g) {
        int   m  = g + 8 * hi;
        float gv = acc[g];
        Ms[(size_t)(i0 + m) * HT + (j0 + l15)] = ss * gv * gv;
    }
}

// ---------------------------------------------------------------------------
// r[j] = sign_j * ||W_j||^2,  and Wt[d][j] = W[j][d] (transpose for the
// F-GEMM B operand).  Grid covers exactly 768 threads.
// ---------------------------------------------------------------------------
__global__ void k_prep(const float* __restrict__ A, const float* __restrict__ B,
                       float* __restrict__ r, float* __restrict__ Wt) {
    int j = blockIdx.x * blockDim.x + threadIdx.x;   // 0..767
    const float* w = wall_row(A, B, j);
    float s = 0.f;
#pragma unroll 16
    for (int k = 0; k < DDIM; ++k) {
        float v = w[k];
        s = fmaf(v, v, s);
        Wt[(size_t)k * HT + j] = v;   // consecutive j across lanes -> coalesced
    }
    r[j] = (j < AROWS) ? s : -s;
}

// ---------------------------------------------------------------------------
// Z = S * W^T (4096x768, K=128), fused Hermite: Hs = sign*h(Z), DH = h'(Z).
// 8 waves/block, one 16x16 tile per wave; grid covers exactly 256*48 tiles.
// ---------------------------------------------------------------------------
__global__ void k_zhd(const float* __restrict__ S, const float* __restrict__ A,
                      const float* __restrict__ B, const float* __restrict__ c,
                      float* __restrict__ Hs, float* __restrict__ DH) {
    const int lane = threadIdx.x & 31;
    const int l15  = lane & 15;
    const int hi   = lane >> 4;
    const int tile = blockIdx.x * 8 + (threadIdx.x >> 5);
    const int bt = tile / 48, jt = tile % 48;
    const int b0 = bt * 16, j0 = jt * 16;
    const float* ra = S + (size_t)(b0 + l15) * DDIM;
    const float* rb = wall_row(A, B, j0 + l15);
    v8f acc = {};
#pragma unroll 8
    for (int k0 = 0; k0 < DDIM; k0 += 4) {
        v2f a = *reinterpret_cast<const v2f*>(ra + k0 + 2 * hi);
        v2f b = *reinterpret_cast<const v2f*>(rb + k0 + 2 * hi);
        acc = wmma_f32(a, b, acc);
    }
    const float c0 = c[0], c1 = c[1], c2 = c[2], c3 = c[3], c4 = c[4], c5 = c[5];
    const float sj = (j0 < AROWS) ? 1.f : -1.f;
#pragma unroll
    for (int g = 0; g < 8; ++g) {
        int   m   = g + 8 * hi;
        float z   = acc[g];
        float he2 = z * z - 1.f;                 // He2
        float he3 = z * he2 - 2.f * z;           // He3
        float he4 = z * he3 - 3.f * he2;         // He4
        float he5 = z * he4 - 4.f * he3;         // He5
        float h   = c0 + c1 * z + c2 * he2 + c3 * he3 + c4 * he4 + c5 * he5;
        float dh  = c1 + 2.f * c2 * z + 3.f * c3 * he2 + 4.f * c4 * he3 + 5.f * c5 * he4;
        size_t idx = (size_t)(b0 + m) * HT + (j0 + l15);
        Hs[idx] = sj * h;
        DH[idx] = dh;
    }
}

// ---------------------------------------------------------------------------
// F = Hs * W + bF   (4096x128, K=768). 8 waves/block, wave = 16-col tile.
// B fragments come from Wt (transposed weights) -> contiguous float2 loads.
// ---------------------------------------------------------------------------
__global__ void k_f(const float* __restrict__ Hs, const float* __restrict__ Wt,
                    const float* __restrict__ bF, float* __restrict__ Fout) {
    const int lane = threadIdx.x & 31;
    const int l15  = lane & 15;
    const int hi   = lane >> 4;
    const int b0   = blockIdx.x * 16;
    const int n0   = (threadIdx.x >> 5) * 16;
    const float* ra = Hs + (size_t)(b0 + l15) * HT;
    const float* rb = Wt + (size_t)(n0 + l15) * HT;  // B[k][n] = W[k][n0+n] = Wt[n0+n][k]
    v8f acc = {};
#pragma unroll 8
    for (int k0 = 0; k0 < HT; k0 += 4) {
        v2f a = *reinterpret_cast<const v2f*>(ra + k0 + 2 * hi);
        v2f b = *reinterpret_cast<const v2f*>(rb + k0 + 2 * hi);
        acc = wmma_f32(a, b, acc);
    }
    float bias = bF[n0 + l15];
#pragma unroll
    for (int g = 0; g < 8; ++g) {
        int m = g + 8 * hi;
        Fout[(size_t)(b0 + m) * DDIM + n0 + l15] = acc[g] + bias;
    }
}

// ---------------------------------------------------------------------------
// Summary: fro2_b = dh_b . M . dh_b (GEMM K=768; M symmetric so
// B[k][n] = Ms[j0+n][k] -> contiguous float2 loads; rowdot folded via LDS
// ds_add_f32 atomics), trace_b = dh_b . r.
// One block (8 waves) per 16 batch rows; each wave owns 6 of the 48 j-tiles.
// ---------------------------------------------------------------------------
__global__ void k_j(const float* __restrict__ DH, const float* __restrict__ Ms,
                    const float* __restrict__ r, float* __restrict__ out2) {
    __shared__ float fro[16];
    const int lane = threadIdx.x & 31;
    const int l15  = lane & 15;
    const int hi   = lane >> 4;
    const int wave = threadIdx.x >> 5;
    const int b0   = blockIdx.x * 16;
    if (threadIdx.x < 16) fro[threadIdx.x] = 0.f;
    __syncthreads();
    const float* ra = DH + (size_t)(b0 + l15) * HT;
    for (int t = 0; t < 6; ++t) {
        const int j0 = (wave * 6 + t) * 16;
        const float* rb = Ms + (size_t)(j0 + l15) * HT;  // symmetry: Ms[k][j] = Ms[j][k]
        v8f acc = {};
#pragma unroll 8
        for (int k0 = 0; k0 < HT; k0 += 4) {
            v2f a = *reinterpret_cast<const v2f*>(ra + k0 + 2 * hi);
            v2f b = *reinterpret_cast<const v2f*>(rb + k0 + 2 * hi);
            acc = wmma_f32(a, b, acc);
        }
#pragma unroll
        for (int g = 0; g < 8; ++g) {
            int   m = g + 8 * hi;
            float v = acc[g] * DH[(size_t)(b0 + m) * HT + (j0 + l15)];
            atomicAdd(&fro[m], v);   // ds_add_f32
        }
    }
    __syncthreads();
    if (threadIdx.x < 16) {
        int m = threadIdx.x;
        const float* dr = DH + (size_t)(b0 + m) * HT;
        float tr = 0.f;
#pragma unroll 8
        for (int j = 0; j < HT; ++j) tr = fmaf(dr[j], r[j], tr);
        out2[(size_t)(b0 + m) * 2 + 0] = tr;
        out2[(size_t)(b0 + m) * 2 + 1] = sqrtf(fmaxf(fro[m], 0.f));
    }
}

extern "C" void kernel_launch(void* const* d_in, const int* in_sizes, int n_in,
                              void* d_out, int out_size, void* d_ws, size_t ws_size,
                              hipStream_t stream) {
    (void)in_sizes; (void)n_in; (void)out_size; (void)ws_size;
    const float* s  = (const float*)d_in[0];
    const float* Aw = (const float*)d_in[1];   // (2,256,128) -> rows 0..511
    const float* Bw = (const float*)d_in[2];   // (1,256,128) -> rows 512..767
    const float* c  = (const float*)d_in[3];   // 6 Hermite coefficients
    const float* bF = (const float*)d_in[4];   // (128,)

    float* Fout = (float*)d_out;                          // 4096*128
    float* out2 = (float*)d_out + (size_t)BATCH * DDIM;   // 4096*2 (trace, fro)

    char*  ws = (char*)d_ws;
    float* Ms = (float*)ws;                                        // 768*768 f32
    float* r  = (float*)(ws + (size_t)HT * HT * 4);                // 768 f32 (+pad)
    float* Wt = (float*)(ws + (size_t)HT * HT * 4 + 4096);         // 128*768 f32
    float* Hs = (float*)(ws + (size_t)HT * HT * 4 + 4096
                            + (size_t)DDIM * HT * 4);              // 4096*768 f32
    float* DH = Hs + (size_t)BATCH * HT;                           // 4096*768 f32

    k_ms  <<<dim3(48, 48), 32, 0, stream>>>(Aw, Bw, Ms);
    k_prep<<<3, 256, 0, stream>>>(Aw, Bw, r, Wt);
    k_zhd <<<(256 * 48) / 8, 256, 0, stream>>>(s, Aw, Bw, c, Hs, DH);
    k_f   <<<256, 256, 0, stream>>>(Hs, Wt, bF, Fout);
    k_j   <<<256, 256, 0, stream>>>(DH, Ms, r, out2);
}